// DirGCNConv_74861279969844
// MI455X (gfx1250) — compile-verified
//
#include <hip/hip_runtime.h>
#include <hip/hip_bf16.h>
#include <math.h>

#define D 128

typedef __attribute__((ext_vector_type(2))) float v2f;
typedef __attribute__((ext_vector_type(8))) float v8f;

// ---------------------------------------------------------------------------
// 0) zero workspace (degrees + both aggregate buffers)
// ---------------------------------------------------------------------------
__global__ void zero_kernel(float* __restrict__ p, long long n) {
    long long i = (long long)blockIdx.x * blockDim.x + threadIdx.x;
    long long stride = (long long)gridDim.x * blockDim.x;
    for (; i < n; i += stride) p[i] = 0.0f;
}

// ---------------------------------------------------------------------------
// 1) degree counting: out_deg[row]++, in_deg[col]++
// ---------------------------------------------------------------------------
__global__ void degree_kernel(const int* __restrict__ ei,
                              float* __restrict__ outdeg,
                              float* __restrict__ indeg, int E) {
    int e = blockIdx.x * blockDim.x + threadIdx.x;
    if (e < E) {
        atomicAdd(&outdeg[ei[e]], 1.0f);
        atomicAdd(&indeg[ei[E + e]], 1.0f);
    }
}

// ---------------------------------------------------------------------------
// 2) deg -> deg^{-1/2} (0 where deg == 0); outdeg/indeg are contiguous
// ---------------------------------------------------------------------------
__global__ void invsqrt_kernel(float* __restrict__ d, int n) {
    int i = blockIdx.x * blockDim.x + threadIdx.x;
    if (i < n) {
        float v = d[i];
        d[i] = (v > 0.0f) ? (1.0f / sqrtf(v)) : 0.0f;
    }
}

// ---------------------------------------------------------------------------
// 3) edge scatter: one wave32 per edge, 4 features per lane (float4 reads,
//    non-returning f32 atomics -> STOREcnt path, stays resident in 192MB L2)
// ---------------------------------------------------------------------------
__global__ void scatter_kernel(const float* __restrict__ x,
                               const int* __restrict__ ei,
                               const float* __restrict__ dout,
                               const float* __restrict__ din,
                               float* __restrict__ aggf,
                               float* __restrict__ aggb, int E) {
    const int lane = threadIdx.x & 31;
    const long long gtid = (long long)blockIdx.x * blockDim.x + threadIdx.x;
    const int e = (int)(gtid >> 5);
    if (e >= E) return;

    const int row = ei[e];
    const int col = ei[E + e];
    const float w = dout[row] * din[col];

    const int f = lane * 4;
    const float4 xc = *reinterpret_cast<const float4*>(x + (long long)col * D + f);
    const float4 xr = *reinterpret_cast<const float4*>(x + (long long)row * D + f);

    float* pf = aggf + (long long)row * D + f;
    float* pb = aggb + (long long)col * D + f;
    atomicAdd(pf + 0, w * xc.x);
    atomicAdd(pf + 1, w * xc.y);
    atomicAdd(pf + 2, w * xc.z);
    atomicAdd(pf + 3, w * xc.w);
    atomicAdd(pb + 0, w * xr.x);
    atomicAdd(pb + 1, w * xr.y);
    atomicAdd(pb + 2, w * xr.z);
    atomicAdd(pb + 3, w * xr.w);
}

// ---------------------------------------------------------------------------
// 4) out = 0.5*(aggf @ Ws) + 0.5*(aggb @ Wd) + 0.5*(bs + bd)
//    One wave per 16x16 output tile; f32 WMMA 16x16x4, K looped 0..128.
//    Alpha folded into A-operand loads so one accumulator serves both GEMMs.
// ---------------------------------------------------------------------------
__global__ void gemm_wmma_kernel(const float* __restrict__ aggf,
                                 const float* __restrict__ aggb,
                                 const float* __restrict__ Ws,
                                 const float* __restrict__ Wd,
                                 const float* __restrict__ bs,
                                 const float* __restrict__ bd,
                                 float* __restrict__ out, int M) {
    const int lane = threadIdx.x & 31;
    const int wid  = (int)(((long long)blockIdx.x * blockDim.x + threadIdx.x) >> 5);
    const int NT   = D / 16;            // 8 tiles along N
    const int mTile = wid / NT;
    const int nTile = wid % NT;
    if (mTile * 16 >= M) return;

    const int mrow = mTile * 16 + (lane & 15);  // A row for this lane
    const int n    = nTile * 16 + (lane & 15);  // B/C column for this lane
    const int kh   = (lane >> 4) * 2;           // lanes 16-31 hold K+2,K+3

    v8f acc = {};

    const float* af = aggf + (long long)mrow * D + kh;
    #pragma unroll 4
    for (int k0 = 0; k0 < D; k0 += 4) {
        v2f a, b;
        a.x = 0.5f * af[k0];
        a.y = 0.5f * af[k0 + 1];
        b.x = Ws[(k0 + kh) * D + n];
        b.y = Ws[(k0 + kh + 1) * D + n];
        acc = __builtin_amdgcn_wmma_f32_16x16x4_f32(false, a, false, b,
                                                    (short)0, acc, false, false);
    }
    const float* ab = aggb + (long long)mrow * D + kh;
    #pragma unroll 4
    for (int k0 = 0; k0 < D; k0 += 4) {
        v2f a, b;
        a.x = 0.5f * ab[k0];
        a.y = 0.5f * ab[k0 + 1];
        b.x = Wd[(k0 + kh) * D + n];
        b.y = Wd[(k0 + kh + 1) * D + n];
        acc = __builtin_amdgcn_wmma_f32_16x16x4_f32(false, a, false, b,
                                                    (short)0, acc, false, false);
    }

    const float bias = 0.5f * (bs[n] + bd[n]);
    // C/D layout: VGPR r holds M = 16*mTile + r + 8*(lane>=16), N = lane&15
    const int mbase = mTile * 16 + (lane >> 4) * 8;
    #pragma unroll
    for (int r = 0; r < 8; ++r) {
        int m = mbase + r;
        if (m < M) out[(long long)m * D + n] = acc[r] + bias;
    }
}

// ---------------------------------------------------------------------------
extern "C" void kernel_launch(void* const* d_in, const int* in_sizes, int n_in,
                              void* d_out, int out_size, void* d_ws, size_t ws_size,
                              hipStream_t stream) {
    const float* x  = (const float*)d_in[0];
    const int*   ei = (const int*)d_in[1];
    const float* Ws = (const float*)d_in[2];
    const float* bs = (const float*)d_in[3];
    const float* Wd = (const float*)d_in[4];
    const float* bd = (const float*)d_in[5];
    float* out = (float*)d_out;

    const int N = in_sizes[0] / D;   // 50000 nodes
    const int E = in_sizes[1] / 2;   // 800000 edges

    // workspace layout: out_deg[N] | in_deg[N] | agg_fwd[N*D] | agg_bwd[N*D]
    float* ws     = (float*)d_ws;
    float* outdeg = ws;
    float* indeg  = outdeg + N;
    float* aggf   = indeg + N;
    float* aggb   = aggf + (long long)N * D;
    const long long ws_elems = 2LL * N + 2LL * N * D;

    zero_kernel<<<2048, 256, 0, stream>>>(ws, ws_elems);

    degree_kernel<<<(E + 255) / 256, 256, 0, stream>>>(ei, outdeg, indeg, E);

    invsqrt_kernel<<<(2 * N + 255) / 256, 256, 0, stream>>>(outdeg, 2 * N);

    {
        long long threads = (long long)E * 32;   // one wave32 per edge
        int blocks = (int)((threads + 255) / 256);
        scatter_kernel<<<blocks, 256, 0, stream>>>(x, ei, outdeg, indeg, aggf, aggb, E);
    }
    {
        int mTiles = (N + 15) / 16;
        int waves  = mTiles * (D / 16);          // one wave per 16x16 tile
        int blocks = (waves * 32 + 255) / 256;
        gemm_wmma_kernel<<<blocks, 256, 0, stream>>>(aggf, aggb, Ws, Wd, bs, bd, out, N);
    }
}